// LeastSquaresContainer_45595372814860
// MI455X (gfx1250) — compile-verified
//
#include <hip/hip_runtime.h>
#include <hip/hip_bf16.h>

// loss = sum_{i,j} (s_i - b_j)^2  with s = A @ x
//      = N * sum(s^2) - 2 * sum(s) * sum(b) + N * sum(b^2)
//
// s computed with V_WMMA_F32_16X16X4_F32: per 16-row tile of A, the B fragment
// broadcasts x[k..k+3] across all 16 columns, so every D column == s tile.
// K = 20 -> 5 chained WMMAs. Each s row is replicated across 16 lanes, so the
// wave-summed moments are scaled by exactly 1/16.

typedef __attribute__((ext_vector_type(2))) float v2f;
typedef __attribute__((ext_vector_type(8))) float v8f;

#define NFEAT   20
#define KCHUNKS 5          // 20 / 4
#define NB      32         // blocks
#define NT      256        // threads per block (8 waves)
#define WPB     (NT / 32)  // waves per block

__global__ __launch_bounds__(NT)
void lsq_partial_kernel(const float* __restrict__ A,
                        const float* __restrict__ b,
                        const float* __restrict__ x,
                        double* __restrict__ ws,
                        int n_rows)
{
    const int lane  = threadIdx.x & 31;
    const int wave  = threadIdx.x >> 5;
    const int gwave = blockIdx.x * WPB + wave;
    const int nwav  = gridDim.x * WPB;

    __shared__ float  xs[32];
    __shared__ double smem[4][WPB];

    if (threadIdx.x < 32)
        xs[threadIdx.x] = (threadIdx.x < NFEAT) ? x[threadIdx.x] : 0.0f;
    __syncthreads();

    // A fragment mapping (32-bit A 16x4): lanes 0-15 hold K {0,1} of rows 0-15,
    // lanes 16-31 hold K {2,3} of the same rows.
    const int row_in_tile = lane & 15;
    const int kbase       = (lane >> 4) * 2;   // 0 or 2

    // B fragment: B[k][j] = x[k] for every column j (broadcast over columns)
    v2f bfrag[KCHUNKS];
#pragma unroll
    for (int c = 0; c < KCHUNKS; ++c) {
        const int k = c * 4 + kbase;
        bfrag[c].x = xs[k];
        bfrag[c].y = xs[k + 1];
    }

    double acc_s = 0.0, acc_s2 = 0.0, acc_b = 0.0, acc_b2 = 0.0;

    // A viewed as float2: row stride = 10, chunk stride = 2, half-lane offset = kbase/2
    const v2f* __restrict__ A2 = (const v2f*)A;
    const int ntiles = (n_rows + 15) >> 4;

    for (int t = gwave; t < ntiles; t += nwav) {   // uniform per wave: EXEC stays full
        const int row   = t * 16 + row_in_tile;
        const bool in   = (row < n_rows);
        const int row_c = in ? row : (n_rows - 1);           // clamp: always legal address
        const size_t base2 = (size_t)row_c * (NFEAT / 2) + (kbase >> 1);

        // Batch all chunk loads (independent b64 loads -> one clause), then select-zero tail
        v2f a[KCHUNKS];
#pragma unroll
        for (int c = 0; c < KCHUNKS; ++c)
            a[c] = A2[base2 + c * 2];
#pragma unroll
        for (int c = 0; c < KCHUNKS; ++c) {
            a[c].x = in ? a[c].x : 0.0f;   // v_cndmask, no EXEC divergence
            a[c].y = in ? a[c].y : 0.0f;
        }

        v8f acc = {};
#pragma unroll
        for (int c = 0; c < KCHUNKS; ++c) {
            // D = A(16x4) * B(4x16) + C  -> every column of D holds the s tile
            acc = __builtin_amdgcn_wmma_f32_16x16x4_f32(
                      false, a[c], false, bfrag[c], (short)0, acc, false, false);
        }

        float ls = 0.0f, ls2 = 0.0f;
#pragma unroll
        for (int r = 0; r < 8; ++r) { const float v = acc[r]; ls += v; ls2 += v * v; }
        acc_s  += (double)ls;
        acc_s2 += (double)ls2;
    }

    // b moments: plain grid-stride
    for (int i = blockIdx.x * NT + threadIdx.x; i < n_rows; i += gridDim.x * NT) {
        const double bv = (double)b[i];
        acc_b  += bv;
        acc_b2 += bv * bv;
    }

    // wave32 tree reduction
#pragma unroll
    for (int off = 16; off > 0; off >>= 1) {
        acc_s  += __shfl_xor(acc_s,  off, 32);
        acc_s2 += __shfl_xor(acc_s2, off, 32);
        acc_b  += __shfl_xor(acc_b,  off, 32);
        acc_b2 += __shfl_xor(acc_b2, off, 32);
    }

    if (lane == 0) {
        smem[0][wave] = acc_s  * (1.0 / 16.0);   // undo 16-lane replication (exact)
        smem[1][wave] = acc_s2 * (1.0 / 16.0);
        smem[2][wave] = acc_b;
        smem[3][wave] = acc_b2;
    }
    __syncthreads();

    if (threadIdx.x == 0) {
        double p0 = 0, p1 = 0, p2 = 0, p3 = 0;
#pragma unroll
        for (int w = 0; w < WPB; ++w) {
            p0 += smem[0][w]; p1 += smem[1][w]; p2 += smem[2][w]; p3 += smem[3][w];
        }
        ws[blockIdx.x * 4 + 0] = p0;
        ws[blockIdx.x * 4 + 1] = p1;
        ws[blockIdx.x * 4 + 2] = p2;
        ws[blockIdx.x * 4 + 3] = p3;
    }
}

__global__ void lsq_final_kernel(const double* __restrict__ ws,
                                 float* __restrict__ out,
                                 int nblocks, int n_rows)
{
    if (threadIdx.x == 0) {
        double Ss = 0, Ss2 = 0, Sb = 0, Sb2 = 0;
        for (int i = 0; i < nblocks; ++i) {
            Ss  += ws[4 * i + 0];
            Ss2 += ws[4 * i + 1];
            Sb  += ws[4 * i + 2];
            Sb2 += ws[4 * i + 3];
        }
        const double N = (double)n_rows;
        out[0] = (float)(N * Ss2 - 2.0 * Ss * Sb + N * Sb2);
    }
}

extern "C" void kernel_launch(void* const* d_in, const int* in_sizes, int n_in,
                              void* d_out, int out_size, void* d_ws, size_t ws_size,
                              hipStream_t stream)
{
    const float* A = (const float*)d_in[0];
    const float* b = (const float*)d_in[1];
    const float* x = (const float*)d_in[2];
    float* out = (float*)d_out;
    double* ws = (double*)d_ws;
    const int n_rows = in_sizes[1];   // 5000

    lsq_partial_kernel<<<NB, NT, 0, stream>>>(A, b, x, ws, n_rows);
    lsq_final_kernel<<<1, 32, 0, stream>>>(ws, out, NB, n_rows);
}